// GraphConv_22067541967338
// MI455X (gfx1250) — compile-verified
//
#include <hip/hip_runtime.h>
#include <hip/hip_bf16.h>

// MI455X (gfx1250) implementation. Memory-bound problem (~1 GB traffic @ 23.3 TB/s,
// ~45us floor), only ~15 GFLOP of GEMM -> fp32 V_WMMA_F32_16X16X4_F32 keeps full
// precision for free while using the matrix pipe. LDS tiles are XOR-swizzled at 16B
// granularity so wave32 ds_load_b64 A-fragment reads cover all 64 banks conflict-free.

#define D_DIM 128
#define H_DIM 256
#define EPSV  1e-5f
#define SLOPE 0.01f
#define WAVES 4   // waves (16-row tiles) per block; LDS = 64KB exactly

typedef __attribute__((ext_vector_type(2))) float v2f;
typedef __attribute__((ext_vector_type(8))) float v8f;

__device__ __forceinline__ v8f wmma4(v2f a, v2f b, v8f c) {
  // D(16x16,f32) += A(16x4,f32) * B(4x16,f32); exact fp32 MACs.
  return __builtin_amdgcn_wmma_f32_16x16x4_f32(false, a, false, b, (short)0, c,
                                               false, false);
}

// XOR-swizzle of LDS column index at 16B (4-float) granularity, keyed by row:
// spreads half-wave A-fragment ds_load_b64's across all 64 LDS banks.
__device__ __forceinline__ int swz(int row, int col) {
  return (((col >> 2) ^ (row & 15)) << 2) | (col & 3);
}

// Order-preserving float <-> uint encoding so float-max becomes atomicMax(u32).
__device__ __forceinline__ unsigned encf(float f) {
  unsigned b = __float_as_uint(f);
  return (b & 0x80000000u) ? ~b : (b | 0x80000000u);
}
__device__ __forceinline__ float decf(unsigned e) {
  unsigned b = (e & 0x80000000u) ? (e & 0x7fffffffu) : ~e;
  return __uint_as_float(b);
}
#define ENC_NEG_INF 0x007FFFFFu  // encf(-inf)

__device__ __forceinline__ float fixfin(float f) {  // !isfinite -> 0 (empty segments)
  return ((__float_as_uint(f) & 0x7f800000u) == 0x7f800000u) ? 0.f : f;
}
__device__ __forceinline__ float sigm(float t) { return 1.f / (1.f + __expf(-t)); }
__device__ __forceinline__ float lrelu(float y) { return (y >= 0.f) ? y : SLOPE * y; }

// ---------------------------------------------------------------- init
__global__ void k_init(float* aggsum, unsigned* aggmax, float* stats, int nnd) {
  int i = blockIdx.x * blockDim.x + threadIdx.x;
  if (i < nnd) { aggsum[i] = 0.f; aggmax[i] = ENC_NEG_INF; }
  if (i < 768) stats[i] = 0.f;  // sum1[256] sq1[256] sum2[128] sq2[128]
}

// ---------------------------------------------------------------- scatter
// one wave32 per edge; lane moves float4 (32 lanes x 16B = 512B row)
__global__ void k_scatter(const float* __restrict__ x,
                          const long long* __restrict__ edges,
                          float* aggsum, unsigned* aggmax, int nedges) {
  int gid  = blockIdx.x * blockDim.x + threadIdx.x;
  int e    = gid >> 5;
  int lane = gid & 31;
  if (e >= nedges) return;
  long s = (long)edges[e];
  long d = (long)edges[(long)nedges + e];
  float4 v = *(const float4*)(x + s * D_DIM + lane * 4);
  float*    ps = aggsum + d * D_DIM + lane * 4;
  unsigned* pm = aggmax + d * D_DIM + lane * 4;
  atomicAdd(ps + 0, v.x); atomicAdd(ps + 1, v.y);
  atomicAdd(ps + 2, v.z); atomicAdd(ps + 3, v.w);
  atomicMax(pm + 0, encf(v.x)); atomicMax(pm + 1, encf(v.y));
  atomicMax(pm + 2, encf(v.z)); atomicMax(pm + 3, encf(v.w));
}

// ---------------------------------------------------------------- merge + GRU
__global__ void __launch_bounds__(WAVES * 32)
k_merge_gru(const float* __restrict__ aggsum, const unsigned* __restrict__ aggmax,
            const float* __restrict__ x,
            const float* __restrict__ mW, const float* __restrict__ mb,
            const float* __restrict__ Wih, const float* __restrict__ Whh,
            const float* __restrict__ bih, const float* __restrict__ bhh,
            float* __restrict__ hout, int n) {
  __shared__ float B0[WAVES][16 * D_DIM];
  __shared__ float B1[WAVES][16 * D_DIM];
  const int w    = threadIdx.x >> 5;
  const int lane = threadIdx.x & 31;
  const int nl   = lane & 15;
  const int hw   = lane >> 4;
  const int row0 = (blockIdx.x * WAVES + w) * 16;
  const v8f zero8 = {0.f,0.f,0.f,0.f,0.f,0.f,0.f,0.f};

  // stage: B0 = agg_sum tile, B1 = decoded agg_max tile (zeros past n)
  for (int m = 0; m < 16; ++m) {
    int row = row0 + m;
    float4 s4 = {0.f,0.f,0.f,0.f}, m4 = {0.f,0.f,0.f,0.f};
    if (row < n) {
      s4 = *(const float4*)(aggsum + (long)row * D_DIM + lane * 4);
      uint4 e4 = *(const uint4*)(aggmax + (long)row * D_DIM + lane * 4);
      m4.x = fixfin(decf(e4.x)); m4.y = fixfin(decf(e4.y));
      m4.z = fixfin(decf(e4.z)); m4.w = fixfin(decf(e4.w));
    }
    int sc = swz(m, lane * 4);
    *(float4*)&B0[w][m * D_DIM + sc] = s4;
    *(float4*)&B1[w][m * D_DIM + sc] = m4;
  }
  __syncthreads();

  // merge GEMM: AGG(16x128) = [S|M](16x256) @ mW(256x128); 8 live accumulators
  v8f acc[8];
#pragma unroll
  for (int c = 0; c < 8; ++c) acc[c] = zero8;
  for (int k0 = 0; k0 < 2 * D_DIM; k0 += 4) {
    int kk = k0 + 2 * hw;
    const float* ap = (kk < D_DIM) ? &B0[w][nl * D_DIM + swz(nl, kk)]
                                   : &B1[w][nl * D_DIM + swz(nl, kk - D_DIM)];
    v2f a = *(const v2f*)ap;
#pragma unroll
    for (int c = 0; c < 8; ++c) {
      int col = c * 16 + nl;
      v2f b;
      b.x = mW[kk * D_DIM + col];
      b.y = mW[(kk + 1) * D_DIM + col];
      acc[c] = wmma4(a, b, acc[c]);
    }
  }
  __syncthreads();
  // write AGG (+bias) into B0; load x tile into B1
#pragma unroll
  for (int c = 0; c < 8; ++c) {
    int col = c * 16 + nl;
    float bias = mb[col];
#pragma unroll
    for (int v = 0; v < 8; ++v) {
      int m = v + 8 * hw;
      B0[w][m * D_DIM + swz(m, col)] = acc[c][v] + bias;
    }
  }
  for (int m = 0; m < 16; ++m) {
    int row = row0 + m;
    float4 x4 = {0.f,0.f,0.f,0.f};
    if (row < n) x4 = *(const float4*)(x + (long)row * D_DIM + lane * 4);
    *(float4*)&B1[w][m * D_DIM + swz(m, lane * 4)] = x4;
  }
  __syncthreads();

  // GRU: per 16-col gate chunk, 6 interleaved WMMA accumulations (K=128).
  // B-frags of Wih/Whh are contiguous 8B loads (gi = agg @ Wih^T layout).
  for (int j = 0; j < 8; ++j) {
    int g = j * 16 + nl;
    v8f ir = zero8, iz = zero8, inn = zero8, hr = zero8, hz = zero8, hn = zero8;
    for (int k0 = 0; k0 < D_DIM; k0 += 4) {
      int kk = k0 + 2 * hw;
      int sa = nl * D_DIM + swz(nl, kk);
      v2f aA = *(const v2f*)&B0[w][sa];
      v2f aX = *(const v2f*)&B1[w][sa];
      v2f br = *(const v2f*)(Wih + (long)(g)        * D_DIM + kk);
      v2f bz = *(const v2f*)(Wih + (long)(g + 128)  * D_DIM + kk);
      v2f bn = *(const v2f*)(Wih + (long)(g + 256)  * D_DIM + kk);
      v2f cr = *(const v2f*)(Whh + (long)(g)        * D_DIM + kk);
      v2f cz = *(const v2f*)(Whh + (long)(g + 128)  * D_DIM + kk);
      v2f cn = *(const v2f*)(Whh + (long)(g + 256)  * D_DIM + kk);
      ir  = wmma4(aA, br, ir);
      iz  = wmma4(aA, bz, iz);
      inn = wmma4(aA, bn, inn);
      hr  = wmma4(aX, cr, hr);
      hz  = wmma4(aX, cz, hz);
      hn  = wmma4(aX, cn, hn);
    }
    float bir = bih[g], biz = bih[g + 128], bin_ = bih[g + 256];
    float bhr = bhh[g], bhz = bhh[g + 128], bhn  = bhh[g + 256];
#pragma unroll
    for (int v = 0; v < 8; ++v) {
      int m = v + 8 * hw;
      int row = row0 + m;
      float xv   = B1[w][m * D_DIM + swz(m, g)];
      float r    = sigm(ir[v] + bir + hr[v] + bhr);
      float z    = sigm(iz[v] + biz + hz[v] + bhz);
      float cand = tanhf(inn[v] + bin_ + r * (hn[v] + bhn));
      float hval = (1.f - z) * cand + z * xv;
      if (row < n) hout[(long)row * D_DIM + g] = hval;
    }
  }
}

// ---------------------------------------------------------------- mlp1 (128->256) + BN1 stats
__global__ void __launch_bounds__(WAVES * 32)
k_mlp1(const float* __restrict__ h, const float* __restrict__ W1,
       const float* __restrict__ b1, float* __restrict__ t1,
       float* stat_sum, float* stat_sq, int n) {
  __shared__ float B0[WAVES][16 * D_DIM];
  const int w    = threadIdx.x >> 5;
  const int lane = threadIdx.x & 31;
  const int nl   = lane & 15;
  const int hw   = lane >> 4;
  const int row0 = (blockIdx.x * WAVES + w) * 16;
  const v8f zero8 = {0.f,0.f,0.f,0.f,0.f,0.f,0.f,0.f};

  for (int m = 0; m < 16; ++m) {
    int row = row0 + m;
    float4 h4 = {0.f,0.f,0.f,0.f};
    if (row < n) h4 = *(const float4*)(h + (long)row * D_DIM + lane * 4);
    *(float4*)&B0[w][m * D_DIM + swz(m, lane * 4)] = h4;
  }
  __syncthreads();

  for (int j = 0; j < 16; ++j) {
    int col = j * 16 + nl;
    v8f acc = zero8;
    for (int k0 = 0; k0 < D_DIM; k0 += 4) {
      int kk = k0 + 2 * hw;
      v2f a = *(const v2f*)&B0[w][nl * D_DIM + swz(nl, kk)];
      v2f b;
      b.x = W1[(long)kk * H_DIM + col];
      b.y = W1[(long)(kk + 1) * H_DIM + col];
      acc = wmma4(a, b, acc);
    }
    float bias = b1[col];
    float ls = 0.f, lq = 0.f;
#pragma unroll
    for (int v = 0; v < 8; ++v) {
      int m = v + 8 * hw;
      int row = row0 + m;
      float y = lrelu(acc[v] + bias);
      if (row < n) {
        t1[(long)row * H_DIM + col] = y;
        ls += y; lq += y * y;
      }
    }
    atomicAdd(&stat_sum[col], ls);
    atomicAdd(&stat_sq[col],  lq);
  }
}

// ---------------------------------------------------------------- BN1 + mlp2 (256->128) + BN2 stats
__global__ void __launch_bounds__(WAVES * 32)
k_mlp2(const float* __restrict__ t1,
       const float* __restrict__ g1, const float* __restrict__ be1,
       const float* __restrict__ s1sum, const float* __restrict__ s1sq,
       const float* __restrict__ W2, const float* __restrict__ b2,
       float* __restrict__ outp, float* s2sum, float* s2sq, int n) {
  __shared__ float BB[WAVES][16 * H_DIM];
  const int w    = threadIdx.x >> 5;
  const int lane = threadIdx.x & 31;
  const int nl   = lane & 15;
  const int hw   = lane >> 4;
  const int row0 = (blockIdx.x * WAVES + w) * 16;
  const v8f zero8 = {0.f,0.f,0.f,0.f,0.f,0.f,0.f,0.f};
  const float invn = 1.f / (float)n;

  // stage: BN1-normalized t1 tile (training-mode stats from pass 1)
  for (int part = 0; part < 2; ++part) {
    int c0 = part * 128 + lane * 4;
    float4 sm = *(const float4*)(s1sum + c0);
    float4 sq = *(const float4*)(s1sq + c0);
    float4 ga = *(const float4*)(g1 + c0);
    float4 bb = *(const float4*)(be1 + c0);
    float4 mean, rs;
    mean.x = sm.x * invn; mean.y = sm.y * invn;
    mean.z = sm.z * invn; mean.w = sm.w * invn;
    rs.x = rsqrtf(fmaxf(sq.x * invn - mean.x * mean.x, 0.f) + EPSV);
    rs.y = rsqrtf(fmaxf(sq.y * invn - mean.y * mean.y, 0.f) + EPSV);
    rs.z = rsqrtf(fmaxf(sq.z * invn - mean.z * mean.z, 0.f) + EPSV);
    rs.w = rsqrtf(fmaxf(sq.w * invn - mean.w * mean.w, 0.f) + EPSV);
    for (int m = 0; m < 16; ++m) {
      int row = row0 + m;
      float4 y4 = {0.f,0.f,0.f,0.f};
      if (row < n) {
        float4 t = *(const float4*)(t1 + (long)row * H_DIM + c0);
        y4.x = ga.x * (t.x - mean.x) * rs.x + bb.x;
        y4.y = ga.y * (t.y - mean.y) * rs.y + bb.y;
        y4.z = ga.z * (t.z - mean.z) * rs.z + bb.z;
        y4.w = ga.w * (t.w - mean.w) * rs.w + bb.w;
      }
      *(float4*)&BB[w][m * H_DIM + swz(m, c0)] = y4;
    }
  }
  __syncthreads();

  for (int j = 0; j < 8; ++j) {
    int col = j * 16 + nl;
    v8f acc = zero8;
    for (int k0 = 0; k0 < H_DIM; k0 += 4) {
      int kk = k0 + 2 * hw;
      v2f a = *(const v2f*)&BB[w][nl * H_DIM + swz(nl, kk)];
      v2f b;
      b.x = W2[(long)kk * D_DIM + col];
      b.y = W2[(long)(kk + 1) * D_DIM + col];
      acc = wmma4(a, b, acc);
    }
    float bias = b2[col];
    float ls = 0.f, lq = 0.f;
#pragma unroll
    for (int v = 0; v < 8; ++v) {
      int m = v + 8 * hw;
      int row = row0 + m;
      float y = lrelu(acc[v] + bias);
      if (row < n) {
        outp[(long)row * D_DIM + col] = y;  // pre-BN2 value
        ls += y; lq += y * y;
      }
    }
    atomicAdd(&s2sum[col], ls);
    atomicAdd(&s2sq[col],  lq);
  }
}

// ---------------------------------------------------------------- BN2 in place
__global__ void k_bnfinal(float* outp, const float* s2sum, const float* s2sq,
                          const float* g2, const float* be2, int n) {
  long i = (long)blockIdx.x * blockDim.x + threadIdx.x;
  long total = (long)n * D_DIM;
  if (i >= total) return;
  int c = (int)(i & (D_DIM - 1));
  float invn = 1.f / (float)n;
  float mean = s2sum[c] * invn;
  float var  = fmaxf(s2sq[c] * invn - mean * mean, 0.f);
  float y = outp[i];
  outp[i] = g2[c] * (y - mean) * rsqrtf(var + EPSV) + be2[c];
}

// ---------------------------------------------------------------- launch
extern "C" void kernel_launch(void* const* d_in, const int* in_sizes, int n_in,
                              void* d_out, int out_size, void* d_ws, size_t ws_size,
                              hipStream_t stream) {
  const float*     x    = (const float*)d_in[0];
  const long long* edges= (const long long*)d_in[1];
  const float* mW  = (const float*)d_in[2];
  const float* mb  = (const float*)d_in[3];
  const float* Wih = (const float*)d_in[4];
  const float* Whh = (const float*)d_in[5];
  const float* bih = (const float*)d_in[6];
  const float* bhh = (const float*)d_in[7];
  const float* W1  = (const float*)d_in[8];
  const float* b1  = (const float*)d_in[9];
  const float* g1  = (const float*)d_in[10];
  const float* be1 = (const float*)d_in[11];
  const float* W2  = (const float*)d_in[12];
  const float* b2  = (const float*)d_in[13];
  const float* g2  = (const float*)d_in[14];
  const float* be2 = (const float*)d_in[15];

  const int  n   = in_sizes[0] / D_DIM;
  const int  ne  = in_sizes[1] / 2;
  const long nnd = (long)n * D_DIM;

  float*    ws     = (float*)d_ws;
  float*    aggsum = ws;                          // nnd floats
  unsigned* aggmax = (unsigned*)(ws + nnd);       // nnd u32
  float*    hbuf   = ws + 2 * nnd;                // nnd floats
  float*    t1     = ws + 3 * nnd;                // 2*nnd floats
  float*    stats  = ws + 5 * nnd;                // 768 floats
  float*    s1sum  = stats;
  float*    s1sq   = stats + 256;
  float*    s2sum  = stats + 512;
  float*    s2sq   = stats + 640;

  const int tb = 256;
  k_init<<<(int)((nnd + tb - 1) / tb), tb, 0, stream>>>(aggsum, aggmax, stats, (int)nnd);

  long sthreads = (long)ne * 32;
  k_scatter<<<(int)((sthreads + tb - 1) / tb), tb, 0, stream>>>(x, edges, aggsum, aggmax, ne);

  int tiles  = (n + 15) / 16;
  int blocks = (tiles + WAVES - 1) / WAVES;
  k_merge_gru<<<blocks, WAVES * 32, 0, stream>>>(aggsum, aggmax, x, mW, mb,
                                                 Wih, Whh, bih, bhh, hbuf, n);
  k_mlp1<<<blocks, WAVES * 32, 0, stream>>>(hbuf, W1, b1, t1, s1sum, s1sq, n);
  k_mlp2<<<blocks, WAVES * 32, 0, stream>>>(t1, g1, be1, s1sum, s1sq, W2, b2,
                                            (float*)d_out, s2sum, s2sq, n);
  k_bnfinal<<<(int)((nnd + tb - 1) / tb), tb, 0, stream>>>((float*)d_out, s2sum, s2sq,
                                                           g2, be2, n);
}